// CombInterference1d_46737834115368
// MI455X (gfx1250) — compile-verified
//
#include <hip/hip_runtime.h>
#include <math.h>

// ---------------------------------------------------------------------------
// Fractional-delay IIR comb bank (CombInterference1d) for MI455X (gfx1250).
//
// Parallel decomposition:
//   grid  = 64 blocks = (b,o) pairs         (all the batch/output parallelism)
//   block = 256 threads = 8 waves: lane (k,i), k=0..31 timesteps, i=0..7 input
//           channels. D >= 32 guarantees a 32-sample superstep only reads
//           samples produced before the superstep -> 1 barrier / 32 samples.
//
// LDS: per-(i) private ring of past outputs, RING=384 (>= Dmax+1+32 = 353,
//      proves no read/write slot collision inside a superstep), row stride 385
//      (odd) for bank spread. Input chunk double-buffered in LDS via async
//      global->LDS loads (ASYNCcnt) -- gfx1250 async path.
//
// Ring indices are strength-reduced: they advance by +32 mod 384 per
// superstep, one conditional wrap each, instead of full recomputation.
//
// i-reduction (mean over 8 channels) = 3x wave32 __shfl_xor; lanes i==0 store
// 4 consecutive floats per wave -> coalesced enough (8 MB/pass total).
//
// Workload is dependency/issue bound (0.2 GFLOP, 24 MB traffic): no WMMA use
// case (the i-contraction has time-varying per-lane operands; a 3-step wave32
// shuffle tree is strictly cheaper than marshaling into WMMA layouts).
// ---------------------------------------------------------------------------

#define T_LEN  32768
#define CB     8        // channels: O == I == 8
#define BATCH  8
#define KSTEP  32       // timesteps per superstep (<= min Dint = 32)
#define RING   384      // ring length >= Dmax(320) + 1 + KSTEP, multiple of 32
#define LP     385      // padded row stride (odd -> LDS bank spread)

#if defined(__gfx1250__) && __has_builtin(__builtin_amdgcn_global_load_async_to_lds_b32)
#define HAVE_ASYNC 1
#else
#define HAVE_ASYNC 0
#endif

#if HAVE_ASYNC
// The builtin's params are global-AS int* and LDS-AS int*. Generic->AS casts
// via integer round-trip (low 32 bits of a generic LDS address are the LDS
// byte offset on AMDGPU; global generic == AS1 bitwise).
#define AS1P(p) ((__attribute__((address_space(1))) int*)(unsigned long long)(p))
#define AS3P(p) ((__attribute__((address_space(3))) int*)(unsigned int)(unsigned long long)(p))

__device__ __forceinline__ void wait_async_zero() {
#if __has_builtin(__builtin_amdgcn_s_wait_asynccnt)
  __builtin_amdgcn_s_wait_asynccnt(0);
#else
  asm volatile("s_wait_asynccnt 0" ::: "memory");
#endif
}
#endif

__global__ __launch_bounds__(256) void fiir_pass_kernel(
    const float* __restrict__ in,   // (B, CB, T)
    float* __restrict__ out,        // (B, CB, T)
    const float* __restrict__ f,    // (CB, CB)
    const float* __restrict__ a,    // (CB, CB)
    float sign)
{
  __shared__ float ring[CB * LP];              // past outputs, one row per i
#if HAVE_ASYNC
  __shared__ __align__(16) float stage[2][KSTEP][CB];  // k-major: conflict-free reads
#endif

  const int tid = threadIdx.x;
  const int i   = tid & 7;        // input channel
  const int k   = tid >> 3;       // timestep within superstep, 0..31
  const int o   = blockIdx.x & 7; // output channel
  const int b   = blockIdx.x >> 3;

  const float* __restrict__ inB  = in  + (size_t)b * CB * T_LEN;
  float* __restrict__       outB = out + ((size_t)b * CB + o) * T_LEN;

#if HAVE_ASYNC
  // Prime the first input chunk (overlaps the LDS ring clear below):
  // stage[0][k][i] = in[b,i,k]
  __builtin_amdgcn_global_load_async_to_lds_b32(
      AS1P(inB + (size_t)i * T_LEN + k), AS3P(&stage[0][k][i]), 0, 0);
#endif

  // Per-(o,i) filter constants (same math as reference, float32 throughout).
  const float fv = f[o * CB + i];
  const float Df = 16000.0f / fv;              // fractional delay in samples
  int Di = (int)floorf(Df);
  Di = Di < 1 ? 1 : (Di > 320 ? 320 : Di);     // clip(,1,L-2) as in reference
  const float fr = Df - (float)Di;
  const float sa = sign * a[o * CB + i];

  float* __restrict__ row = &ring[i * LP];

  // Zero-init ring (reference buffer starts at zero; unwritten slots read 0).
  for (int idx = tid; idx < CB * LP; idx += 256) ring[idx] = 0.0f;

#if HAVE_ASYNC
  wait_async_zero();
#endif
  __syncthreads();

  // Strength-reduced ring cursors (all advance +KSTEP mod RING per superstep).
  int wIdx = k;                 // write slot:  (n0 + k) mod RING
  int r1   = k - Di + RING;     // tap y[n-Di]: k-Di in [-320,-1] -> [64,383]
  int r2   = r1 - 1;            // tap y[n-Di-1]

  for (int n0 = 0; n0 < T_LEN; n0 += KSTEP) {
#if HAVE_ASYNC
    const int cur = (n0 >> 5) & 1;
    // Prefetch next chunk into the other staging buffer (async, ASYNCcnt).
    if (n0 + KSTEP < T_LEN) {
      __builtin_amdgcn_global_load_async_to_lds_b32(
          AS1P(inB + (size_t)i * T_LEN + (n0 + KSTEP) + k),
          AS3P(&stage[cur ^ 1][k][i]), 0, 0);
    }
#else
    if (n0 + KSTEP < T_LEN)
      __builtin_prefetch(inB + (size_t)i * T_LEN + (n0 + KSTEP) + k, 0, 0);
#endif

    // Fractional-delay taps: y[n-Di], y[n-Di-1]. Di >= 32 = KSTEP, so taps
    // come strictly from earlier supersteps (barrier-ordered).
    const float d1 = row[r1];
    const float d2 = row[r2];
    const float del = (1.0f - fr) * d1 + fr * d2;

#if HAVE_ASYNC
    const float xin = stage[cur][k][i];
#else
    const float xin = inB[(size_t)i * T_LEN + n0 + k];
#endif

    const float y = fmaf(sa, del, xin);
    row[wIdx] = y;

    // mean over i: 8-lane xor-shuffle tree inside each wave32.
    float s = y;
    s += __shfl_xor(s, 1, 32);
    s += __shfl_xor(s, 2, 32);
    s += __shfl_xor(s, 4, 32);
    if (i == 0) outB[n0 + k] = s * 0.125f;

    // Advance cursors: +KSTEP mod RING, one conditional wrap each.
    wIdx += KSTEP; if (wIdx >= RING) wIdx -= RING;
    r1   += KSTEP; if (r1   >= RING) r1   -= RING;
    r2   += KSTEP; if (r2   >= RING) r2   -= RING;

#if HAVE_ASYNC
    wait_async_zero();   // staged chunk resident before the barrier publishes it
#endif
    __syncthreads();     // publish this superstep's ring writes (+ staged chunk)
  }
}

extern "C" void kernel_launch(void* const* d_in, const int* in_sizes, int n_in,
                              void* d_out, int out_size, void* d_ws, size_t ws_size,
                              hipStream_t stream) {
  (void)in_sizes; (void)n_in; (void)out_size; (void)ws_size;
  const float* x = (const float*)d_in[0];   // (8, 8, 32768) f32
  const float* f = (const float*)d_in[1];   // (8, 8) f32
  const float* a = (const float*)d_in[2];   // (8, 8) f32
  float* y1  = (float*)d_ws;                // intermediate (8,8,32768) = 8 MB
  float* out = (float*)d_out;

  dim3 grid(BATCH * CB);   // 64 blocks: one (b,o) group per WGP
  dim3 block(256);         // 8 wave32

  // Pass 1: anti-comb (sign = -1), x -> ws. Pass 2: comb (sign = +1), ws -> out.
  fiir_pass_kernel<<<grid, block, 0, stream>>>(x,  y1,  f, a, -1.0f);
  fiir_pass_kernel<<<grid, block, 0, stream>>>(y1, out, f, a,  1.0f);
}